// SEGNN_81844896793188
// MI455X (gfx1250) — compile-verified
//
#include <hip/hip_runtime.h>

// ---------------------------------------------------------------------------
// SEGNN forward for MI455X (gfx1250, wave32).
// Every bilinear tensor product  out[n,k] = sum_{i,j} f[n,i]*a[n,j]*W[i,j,k]
// is a GEMM with K = KIN*8 (z[n, i*8+j] = f[n,i]*a[n,j], W flat [KIN*8, 64])
// executed on the fp32 matrix pipe with V_WMMA_F32_16X16X4_F32.
// Row counts (10000 nodes, 160000 edges) are multiples of 16, so every
// 16-row WMMA tile is full: no bounds guards anywhere in the hot path.
// Weights are repacked once per launch into (k,k+1)-pair-interleaved layout
// so each lane's B operand is one global_load_b64.
// ---------------------------------------------------------------------------

typedef float v2f __attribute__((ext_vector_type(2)));
typedef float v8f __attribute__((ext_vector_type(8)));

static constexpr int N_NODES = 10000;   // multiple of 16
static constexpr int N_EDGES = 160000;  // multiple of 16
static constexpr int NB      = 16;
static constexpr int F_IN    = 16;
static constexpr int H       = 64;
static constexpr int NATT    = 8;
static constexpr int NLAYER  = 2;
static constexpr int MSG_IN  = 2 * (H + 1) + 1; // 131
static constexpr int UPD_IN  = H + 1 + H;       // 129

enum Mode { M_EMB, M_MSG1, M_MSG2, M_UPD1, M_UPD2, M_PRE1, M_PRE2 };

static __device__ __forceinline__ float siluf(float x) {
  return x * (1.0f / (1.0f + __expf(-x)));
}

// Wave-private LDS region: only intra-wave ordering needed (DS is in-order
// per wave); barriers stop compiler reordering across the fence.
static __device__ __forceinline__ void wave_lds_fence() {
  __builtin_amdgcn_wave_barrier();
  asm volatile("s_wait_dscnt 0" ::: "memory");
  __builtin_amdgcn_wave_barrier();
}

static __device__ __forceinline__ void atom_add_f32(float* p, float v) {
  unsafeAtomicAdd(p, v); // native global_atomic_add_f32
}

// ---------------------------------------------------------------------------
// Repack W[k][n] (k < ktot, n < 64) -> Wp[k/2][n] = (W[k][n], W[k+1][n])
// so the B operand per lane is one 8-byte load.
// ---------------------------------------------------------------------------
__global__ void repack_kernel(const float* __restrict__ W,
                              float* __restrict__ Wp, int ktot) {
  int i = blockIdx.x * blockDim.x + threadIdx.x;
  if (i < ktot * H) {
    int k = i >> 6, n = i & 63;
    Wp[(((size_t)(k >> 1) * H) + n) * 2 + (k & 1)] = W[i];
  }
}

// ---------------------------------------------------------------------------
// Generic tensor-product GEMM: [rows x KIN*8] (implicit z) @ W[KIN*8 x 64]
// One wave per 16-row tile, 4 waves / block, grid-stride over tiles.
// ---------------------------------------------------------------------------
template <int MODE, int KIN>
__global__ __launch_bounds__(128) void tp_kernel(
    const float* __restrict__ featp,  // primary per-row features
    const float* __restrict__ anf,    // additional node features [N]
    const float* __restrict__ amf,    // additional message features [E]
    const float* __restrict__ aggin,  // aggregated messages [N x 64]
    const int*   __restrict__ esrc,   // edge_index[0]
    const int*   __restrict__ edst,   // edge_index[1]
    const float* __restrict__ attrp,  // edge_attr or node_attr [rows x 8]
    const v2f*   __restrict__ Wp,     // repacked weights [KIN*8/2][64] pairs
    const float* __restrict__ bias,   // [64]
    float*       __restrict__ outp,   // dense out, or h for UPD2 residual
    float*       __restrict__ scatp,  // agg for MSG2 atomic scatter
    int rows)                         // multiple of 16
{
  constexpr int KTOT = KIN * NATT;                 // divisible by 4
  constexpr int KP   = (KIN & 1) ? KIN : KIN + 1;  // bank-friendly pad
  __shared__ float s_feat[4][16][KP];
  __shared__ float s_attr[4][16][9];

  const int lane  = threadIdx.x & 31;
  const int wv    = threadIdx.x >> 5;
  const int mrow  = lane & 15;              // M (A op) / N (B op) index
  const int khalf = (lane >> 4) << 1;       // 0 (lanes 0-15) or 2 (16-31)
  const int ntile = rows >> 4;
  const int wstep = gridDim.x * 4;

  for (int tile = blockIdx.x * 4 + wv; tile < ntile; tile += wstep) {
    const int base = tile << 4;

    // ---- stage features + attrs for 16 rows into this wave's LDS region
    for (int r = 0; r < 16; ++r) {
      const int rg = base + r;
      if (lane < NATT) s_attr[wv][r][lane] = attrp[(size_t)rg * NATT + lane];
      int sE = 0, dE = 0;
      if (MODE == M_MSG1) { sE = esrc[rg]; dE = edst[rg]; }
      for (int c = lane; c < KIN; c += 32) {
        float v;
        if (MODE == M_EMB) {
          v = (c < F_IN) ? featp[(size_t)rg * F_IN + c] : anf[rg];
        } else if (MODE == M_MSG1) {
          if      (c < H)          v = featp[(size_t)dE * H + c];
          else if (c == H)         v = anf[dE];
          else if (c < 2 * H + 1)  v = featp[(size_t)sE * H + (c - H - 1)];
          else if (c == 2 * H + 1) v = anf[sE];
          else                     v = amf[rg];
        } else if (MODE == M_UPD1) {
          if      (c < H)          v = featp[(size_t)rg * H + c];
          else if (c == H)         v = anf[rg];
          else                     v = aggin[(size_t)rg * H + (c - H - 1)];
        } else { // MSG2 / UPD2 / PRE1 / PRE2: plain [rows x 64]
          v = featp[(size_t)rg * H + c];
        }
        s_feat[wv][r][c] = v;
      }
    }
    wave_lds_fence();

    // ---- WMMA fp32 GEMM, K in steps of 4, 4 output column tiles (H=64)
    v8f acc0 = {0,0,0,0,0,0,0,0};
    v8f acc1 = {0,0,0,0,0,0,0,0};
    v8f acc2 = {0,0,0,0,0,0,0,0};
    v8f acc3 = {0,0,0,0,0,0,0,0};

    #pragma unroll 4
    for (int k0 = 0; k0 < KTOT; k0 += 4) {
      const int kA = k0 + khalf;            // this lane-half's K pair
      v2f a;
      a.x = s_feat[wv][mrow][(kA)     >> 3] * s_attr[wv][mrow][(kA)     & 7];
      a.y = s_feat[wv][mrow][(kA + 1) >> 3] * s_attr[wv][mrow][(kA + 1) & 7];
      const v2f* wr = Wp + ((size_t)(kA >> 1) * H) + mrow; // one b64 per tile
      acc0 = __builtin_amdgcn_wmma_f32_16x16x4_f32(false, a, false, wr[0],
                                                   (short)0, acc0, false, false);
      acc1 = __builtin_amdgcn_wmma_f32_16x16x4_f32(false, a, false, wr[16],
                                                   (short)0, acc1, false, false);
      acc2 = __builtin_amdgcn_wmma_f32_16x16x4_f32(false, a, false, wr[32],
                                                   (short)0, acc2, false, false);
      acc3 = __builtin_amdgcn_wmma_f32_16x16x4_f32(false, a, false, wr[48],
                                                   (short)0, acc3, false, false);
    }

    // ---- epilogue: bias, activation, store / residual / atomic scatter
    // Full tiles everywhere (rows % 16 == 0): straight-line, no guards.
    v8f accs[4] = {acc0, acc1, acc2, acc3};
    const int rhalf = (lane >> 4) << 3;     // C/D: lanes 16-31 hold M=r+8
    #pragma unroll
    for (int nt = 0; nt < 4; ++nt) {
      const int col = nt * 16 + mrow;
      const float bv = bias[col];
      #pragma unroll
      for (int r = 0; r < 8; ++r) {
        const int rg = base + r + rhalf;
        float v = accs[nt][r] + bv;
        if (MODE == M_MSG1 || MODE == M_MSG2 ||
            MODE == M_UPD1 || MODE == M_PRE1) v = siluf(v);
        if (MODE == M_MSG2) {
          atom_add_f32(&scatp[(size_t)edst[rg] * H + col], v); // segment_sum
        } else if (MODE == M_UPD2) {
          outp[(size_t)rg * H + col] += v;                     // residual
        } else {
          outp[(size_t)rg * H + col] = v;
        }
      }
    }
    wave_lds_fence(); // LDS reads done before next tile overwrites
  }
}

// ---------------------------------------------------------------------------
__global__ void zero_kernel(float* __restrict__ p, int n) {
  int i = blockIdx.x * blockDim.x + threadIdx.x;
  if (i < n) p[i] = 0.0f;
}

__global__ void pool_kernel(const float* __restrict__ p,
                            const int* __restrict__ batch,
                            float* __restrict__ g, float* __restrict__ cnt,
                            int n) {
  int i = blockIdx.x * blockDim.x + threadIdx.x;
  int node = i >> 6, c = i & 63;
  if (node < n) {
    int b = batch[node];
    atom_add_f32(&g[(size_t)b * H + c], p[i]);
    if (c == 0) atom_add_f32(&cnt[b], 1.0f);
  }
}

__global__ __launch_bounds__(64) void head_kernel(
    const float* __restrict__ g, const float* __restrict__ cnt,
    const float* __restrict__ Wq1, const float* __restrict__ bq1,
    const float* __restrict__ Wq2, const float* __restrict__ bq2,
    float* __restrict__ out) {
  __shared__ float gb[64], t[64];
  const int c = threadIdx.x;
  for (int b = 0; b < NB; ++b) {
    float cn = cnt[b]; if (cn < 1.0f) cn = 1.0f;
    gb[c] = g[(size_t)b * H + c] / cn;
    __syncthreads();
    float s = bq1[c];
    for (int k = 0; k < H; ++k) s += gb[k] * Wq1[k * H + c];
    t[c] = siluf(s);
    __syncthreads();
    if (c == 0) {
      float o = bq2[0];
      for (int k = 0; k < H; ++k) o += t[k] * Wq2[k];
      out[b] = o;
    }
    __syncthreads();
  }
}

// ---------------------------------------------------------------------------
extern "C" void kernel_launch(void* const* d_in, const int* in_sizes, int n_in,
                              void* d_out, int out_size, void* d_ws,
                              size_t ws_size, hipStream_t stream) {
  const float* x     = (const float*)d_in[0];
  const int*   ei    = (const int*)  d_in[1];
  const float* eattr = (const float*)d_in[2];
  const float* nattr = (const float*)d_in[3];
  const float* amf   = (const float*)d_in[4];
  const float* anf   = (const float*)d_in[5];
  const int*   batch = (const int*)  d_in[6];
  const float* Wemb  = (const float*)d_in[7];
  const float* bemb  = (const float*)d_in[8];
  const float* Wm1   = (const float*)d_in[9];
  const float* bm1   = (const float*)d_in[10];
  const float* Wm2   = (const float*)d_in[11];
  const float* bm2   = (const float*)d_in[12];
  const float* Wu1   = (const float*)d_in[13];
  const float* bu1   = (const float*)d_in[14];
  const float* Wu2   = (const float*)d_in[15];
  const float* bu2   = (const float*)d_in[16];
  const float* Wp1   = (const float*)d_in[17];
  const float* bp1   = (const float*)d_in[18];
  const float* Wp2   = (const float*)d_in[19];
  const float* bp2   = (const float*)d_in[20];
  const float* Wq1   = (const float*)d_in[21];
  const float* bq1   = (const float*)d_in[22];
  const float* Wq2   = (const float*)d_in[23];
  const float* bq2   = (const float*)d_in[24];
  (void)in_sizes; (void)n_in; (void)out_size; (void)ws_size;

  const int* esrc = ei;            // edge_index[0] -> message source
  const int* edst = ei + N_EDGES;  // edge_index[1] -> aggregation target

  float* ws   = (float*)d_ws;
  float* h    = ws;                               // N x 64
  float* mbuf = h    + (size_t)N_NODES * H;       // E x 64
  float* agg  = mbuf + (size_t)N_EDGES * H;       // N x 64
  float* ut   = agg  + (size_t)N_NODES * H;       // N x 64
  float* pbuf = ut   + (size_t)N_NODES * H;       // N x 64
  float* g    = pbuf + (size_t)N_NODES * H;       // 16 x 64
  float* cnt  = g    + (size_t)NB * H;            // 16   (contiguous after g)
  float* wpk  = cnt  + 64;                        // repacked weights

  // Repacked-weight sub-layout (all K totals, in K-rows of 64 floats)
  const int K_EMB = (F_IN + 1) * NATT;   // 136
  const int K_M1  = MSG_IN * NATT;       // 1048
  const int K_M2  = H * NATT;            // 512
  const int K_U1  = UPD_IN * NATT;       // 1032
  float* pWemb = wpk;
  float* pWm1[NLAYER], *pWm2[NLAYER], *pWu1[NLAYER], *pWu2[NLAYER];
  {
    float* cur = pWemb + (size_t)K_EMB * H;
    for (int l = 0; l < NLAYER; ++l) { pWm1[l] = cur; cur += (size_t)K_M1 * H; }
    for (int l = 0; l < NLAYER; ++l) { pWm2[l] = cur; cur += (size_t)K_M2 * H; }
    for (int l = 0; l < NLAYER; ++l) { pWu1[l] = cur; cur += (size_t)K_U1 * H; }
    for (int l = 0; l < NLAYER; ++l) { pWu2[l] = cur; cur += (size_t)K_M2 * H; }
    float* pWpre1 = cur; cur += (size_t)K_M2 * H;
    float* pWpre2 = cur; cur += (size_t)K_M2 * H;
    // repack everything (cheap: <2 MB total, L2-resident)
    auto rp = [&](const float* src, float* dst, int ktot) {
      repack_kernel<<<(ktot * H + 255) / 256, 256, 0, stream>>>(src, dst, ktot);
    };
    rp(Wemb, pWemb, K_EMB);
    for (int l = 0; l < NLAYER; ++l) {
      rp(Wm1 + (size_t)l * K_M1 * H, pWm1[l], K_M1);
      rp(Wm2 + (size_t)l * K_M2 * H, pWm2[l], K_M2);
      rp(Wu1 + (size_t)l * K_U1 * H, pWu1[l], K_U1);
      rp(Wu2 + (size_t)l * K_M2 * H, pWu2[l], K_M2);
    }
    rp(Wp1, pWpre1, K_M2);
    rp(Wp2, pWpre2, K_M2);

    const dim3 blk(128);
    auto grid_for = [](int rows) -> dim3 {
      int b = ((rows >> 4) + 3) / 4;
      return dim3((unsigned)(b < 2048 ? b : 2048));
    };
    const dim3 gN = grid_for(N_NODES), gE = grid_for(N_EDGES);

    // embedding: h = TP(concat(x, anf), node_attr, W_emb) + b
    tp_kernel<M_EMB, F_IN + 1><<<gN, blk, 0, stream>>>(
        x, anf, nullptr, nullptr, nullptr, nullptr, nattr, (const v2f*)pWemb,
        bemb, h, nullptr, N_NODES);

    for (int l = 0; l < NLAYER; ++l) {
      tp_kernel<M_MSG1, MSG_IN><<<gE, blk, 0, stream>>>(
          h, anf, amf, nullptr, esrc, edst, eattr, (const v2f*)pWm1[l],
          bm1 + l * H, mbuf, nullptr, N_EDGES);

      zero_kernel<<<(N_NODES * H + 255) / 256, 256, 0, stream>>>(agg,
                                                                 N_NODES * H);

      tp_kernel<M_MSG2, H><<<gE, blk, 0, stream>>>(
          mbuf, nullptr, nullptr, nullptr, esrc, edst, eattr,
          (const v2f*)pWm2[l], bm2 + l * H, nullptr, agg, N_EDGES);

      tp_kernel<M_UPD1, UPD_IN><<<gN, blk, 0, stream>>>(
          h, anf, nullptr, agg, nullptr, nullptr, nattr, (const v2f*)pWu1[l],
          bu1 + l * H, ut, nullptr, N_NODES);

      tp_kernel<M_UPD2, H><<<gN, blk, 0, stream>>>(
          ut, nullptr, nullptr, nullptr, nullptr, nullptr, nattr,
          (const v2f*)pWu2[l], bu2 + l * H, h, nullptr, N_NODES);
    }

    // pre-pool MLP
    tp_kernel<M_PRE1, H><<<gN, blk, 0, stream>>>(
        h, nullptr, nullptr, nullptr, nullptr, nullptr, nattr,
        (const v2f*)pWpre1, bp1, ut, nullptr, N_NODES);
    tp_kernel<M_PRE2, H><<<gN, blk, 0, stream>>>(
        ut, nullptr, nullptr, nullptr, nullptr, nullptr, nattr,
        (const v2f*)pWpre2, bp2, pbuf, nullptr, N_NODES);

    // global mean pool + head MLP
    zero_kernel<<<1, 256, 0, stream>>>(g, NB * H + NB); // zeros only g[0..255] (one block of 256 threads)
    pool_kernel<<<(N_NODES * H + 255) / 256, 256, 0, stream>>>(
        pbuf, batch, g, cnt, N_NODES);
    head_kernel<<<1, 64, 0, stream>>>(g, cnt, Wq1, bq1, Wq2, bq2,
                                      (float*)d_out);
  }
}